// ResamplerND_13065290514481
// MI455X (gfx1250) — compile-verified
//
#include <hip/hip_runtime.h>
#include <cstdint>

typedef float v4f __attribute__((ext_vector_type(4)));

namespace {
constexpr int Dd = 64, Hh = 128, Ww = 128, Cc = 16;
constexpr int VOX_PER_BATCH = Dd * Hh * Ww;  // 1 << 20
}

// One voxel per quad of lanes; each lane handles 4 of the 16 channels.
// blockDim = 256 -> 64 voxels per block, 8 wave32s.
__global__ __launch_bounds__(256) void resampler3d_kernel(
    const float* __restrict__ data, const float* __restrict__ warp,
    float* __restrict__ out, int n_vox) {
  __shared__ float s_warp[64 * 3];  // 64 voxels x (z,y,x)

  const int t = threadIdx.x;
  const int gt = blockIdx.x * 256 + t;
  const int vox = gt >> 2;   // global voxel id
  const int q = gt & 3;      // channel quarter (4 floats)
  const int block_vox0 = blockIdx.x * 64;

  // ---- Stage this block's warp coords into LDS with gfx1250 async DMA ----
  // 64 voxels * 3 dwords = 192 dwords; threads 0..191 copy one dword each.
  if (t < 192) {
    const int gdw = block_vox0 * 3 + t;  // dword index into warp[]
    if (gdw < n_vox * 3) {
      const unsigned lds_off = (unsigned)(uintptr_t)&s_warp[t];  // LDS byte addr
      const unsigned voff = (unsigned)gdw * 4u;                  // byte offset
      asm volatile("global_load_async_to_lds_b32 %0, %1, %2"
                   :
                   : "v"(lds_off), "v"(voff), "s"(warp)
                   : "memory");
    }
  }
  asm volatile("s_wait_asynccnt 0" ::: "memory");
  __syncthreads();

  if (vox >= n_vox) return;

  const int lv = t >> 2;  // local voxel within block
  const float wz = s_warp[lv * 3 + 0];
  const float wy = s_warp[lv * 3 + 1];
  const float wx = s_warp[lv * 3 + 2];

  const float fz = floorf(wz), fy = floorf(wy), fx = floorf(wx);
  const float rz = wz - fz, ry = wy - fy, rx = wx - fx;

  const int z0 = (int)fz, y0 = (int)fy, x0 = (int)fx;
  const int z0c = min(max(z0, 0), Dd - 1), z1c = min(max(z0 + 1, 0), Dd - 1);
  const int y0c = min(max(y0, 0), Hh - 1), y1c = min(max(y0 + 1, 0), Hh - 1);
  const int x0c = min(max(x0, 0), Ww - 1), x1c = min(max(x0 + 1, 0), Ww - 1);

  const int b = vox / VOX_PER_BATCH;        // power-of-two -> shift
  const int bb = b * (VOX_PER_BATCH * Cc);  // batch base in floats

  const int r00 = bb + ((z0c * Hh + y0c) * Ww) * Cc;
  const int r01 = bb + ((z0c * Hh + y1c) * Ww) * Cc;
  const int r10 = bb + ((z1c * Hh + y0c) * Ww) * Cc;
  const int r11 = bb + ((z1c * Hh + y1c) * Ww) * Cc;
  const int cq = q * 4;

  const float* p000 = data + r00 + x0c * Cc + cq;
  const float* p001 = data + r00 + x1c * Cc + cq;
  const float* p010 = data + r01 + x0c * Cc + cq;
  const float* p011 = data + r01 + x1c * Cc + cq;
  const float* p100 = data + r10 + x0c * Cc + cq;
  const float* p101 = data + r10 + x1c * Cc + cq;
  const float* p110 = data + r11 + x0c * Cc + cq;
  const float* p111 = data + r11 + x1c * Cc + cq;

  // Kick the 8 gather lines toward L0/L2 while we do the weight VALU work.
  __builtin_prefetch(p000, 0, 3);
  __builtin_prefetch(p001, 0, 3);
  __builtin_prefetch(p010, 0, 3);
  __builtin_prefetch(p011, 0, 3);
  __builtin_prefetch(p100, 0, 3);
  __builtin_prefetch(p101, 0, 3);
  __builtin_prefetch(p110, 0, 3);
  __builtin_prefetch(p111, 0, 3);

  const float mz = 1.0f - rz, my = 1.0f - ry, mx = 1.0f - rx;
  const float w000 = mz * my * mx, w001 = mz * my * rx;
  const float w010 = mz * ry * mx, w011 = mz * ry * rx;
  const float w100 = rz * my * mx, w101 = rz * my * rx;
  const float w110 = rz * ry * mx, w111 = rz * ry * rx;

  v4f acc = w000 * *(const v4f*)p000;
  acc += w001 * *(const v4f*)p001;
  acc += w010 * *(const v4f*)p010;
  acc += w011 * *(const v4f*)p011;
  acc += w100 * *(const v4f*)p100;
  acc += w101 * *(const v4f*)p101;
  acc += w110 * *(const v4f*)p110;
  acc += w111 * *(const v4f*)p111;

  // Write-once output: non-temporal so the gather volume stays L2-resident
  // (data 128MB fits the 192MB L2; data+out would not).
  __builtin_nontemporal_store(acc, (v4f*)(out + (size_t)vox * Cc + cq));
}

extern "C" void kernel_launch(void* const* d_in, const int* in_sizes, int n_in,
                              void* d_out, int out_size, void* d_ws, size_t ws_size,
                              hipStream_t stream) {
  const float* data = (const float*)d_in[0];
  const float* warp = (const float*)d_in[1];
  float* out = (float*)d_out;

  const int n_vox = in_sizes[1] / 3;                 // B*D*H*W
  const long long threads = (long long)n_vox * 4;    // 4 lanes per voxel
  const int blocks = (int)((threads + 255) / 256);

  resampler3d_kernel<<<blocks, 256, 0, stream>>>(data, warp, out, n_vox);
}